// PairwiseConv_83708912599446
// MI455X (gfx1250) — compile-verified
//
#include <hip/hip_runtime.h>

typedef __attribute__((ext_vector_type(2))) float v2f;
typedef __attribute__((ext_vector_type(8))) float v8f;

// Problem constants (from reference)
#define BATCH 4
#define C_IN 64
#define NNODE 4096
#define NEDGE 131072
#define C_OUT 63
#define BC (BATCH * C_IN) // 256

// ---------------- zero scratch ----------------
__global__ void zero_kernel(float* __restrict__ p, int n) {
    int i = blockIdx.x * blockDim.x + threadIdx.x;
    if (i < n) p[i] = 0.0f;
}

// ---------------- x[b][c][n] -> xt[n][b*64+c] ----------------
__global__ void transpose_kernel(const float* __restrict__ x, float* __restrict__ xt) {
    int gid = blockIdx.x * blockDim.x + threadIdx.x; // BC*NNODE threads
    int n  = gid & (NNODE - 1);
    int bc = gid >> 12;
    xt[(size_t)n * BC + bc] = x[(size_t)bc * NNODE + n];
}

// ---------------- pad W (63x64x2) -> Wp0/Wp1 (64x64), bias ----------------
__global__ void padw_kernel(const float* __restrict__ W, const float* __restrict__ bvec,
                            float* __restrict__ Wp0, float* __restrict__ Wp1,
                            float* __restrict__ bias_p) {
    int gid = blockIdx.x * blockDim.x + threadIdx.x; // 4096 threads
    int o = gid >> 6, c = gid & 63;
    float w0 = 0.0f, w1 = 0.0f;
    if (o < C_OUT) {
        w0 = W[o * (C_IN * 2) + c * 2 + 0];
        w1 = W[o * (C_IN * 2) + c * 2 + 1];
    }
    Wp0[gid] = w0;
    Wp1[gid] = w1;
    if (c == 0) bias_p[o] = (o < C_OUT) ? bvec[o] : 0.0f;
}

// ---------------- per-edge stats: adj_sum (by dst), cnt_src (by src) ----------------
__global__ void edge_stats_kernel(const int* __restrict__ idx, const float* __restrict__ vals,
                                  float* __restrict__ adj_sum, float* __restrict__ cnt_src) {
    int e = blockIdx.x * blockDim.x + threadIdx.x;
    if (e >= NEDGE) return;
    int s = idx[e];          // row 0 = src
    int d = idx[NEDGE + e];  // row 1 = dst
    atomicAdd(&adj_sum[d], vals[e]);
    atomicAdd(&cnt_src[s], 1.0f);
}

// ---------------- agg[:,:,src] += x[:,:,dst]  (one wave per edge, 1KB column) ----------------
__global__ __launch_bounds__(256) void aggregate_kernel(const int* __restrict__ idx,
                                                        const float* __restrict__ xt,
                                                        float* __restrict__ agg) {
    int wave = (blockIdx.x * blockDim.x + threadIdx.x) >> 5;
    int lane = threadIdx.x & 31;
    if (wave >= NEDGE) return;
    int s = idx[wave];
    int d = idx[NEDGE + wave];
    const float* srccol = xt + (size_t)d * BC;
    float* dstcol = agg + (size_t)s * BC;
#pragma unroll
    for (int k = 0; k < BC / 32; ++k)
        atomicAdd(&dstcol[lane + 32 * k], srccol[lane + 32 * k]);
}

// ---------------- WMMA GEMM: y = (cnt*(W0·X + bias) + W1·AGG)/denom ----------------
// One wave per (b, otile, ntile). A=16x4 f32 (W rows), B=4x16 f32 (node cols).
__global__ __launch_bounds__(256) void gemm_wmma_kernel(const float* __restrict__ xt,
                                                        const float* __restrict__ agg,
                                                        const float* __restrict__ Wp0,
                                                        const float* __restrict__ Wp1,
                                                        const float* __restrict__ bias_p,
                                                        const float* __restrict__ adj_sum,
                                                        const float* __restrict__ cnt_src,
                                                        float* __restrict__ out) {
    int wave = (blockIdx.x * blockDim.x + threadIdx.x) >> 5;
    int lane = threadIdx.x & 31;
    // 4096 waves total: wave = ((bb*4 + otile) << 8) | ntile
    int ntile = wave & 255;
    int rest  = wave >> 8;
    int otile = rest & 3;
    int bb    = rest >> 2;

    // A-matrix 16x4 layout: lane L holds row M=L%16, K-pair {0,1} (L<16) or {2,3} (L>=16)
    // B-matrix 4x16 layout: lane L holds col N=L%16, same K-pair split
    int MN = lane & 15;
    int k0 = (lane >> 4) << 1; // 0 or 2

    int n = ntile * 16 + MN;
    const float* xcol = xt  + (size_t)n * BC + bb * C_IN; // X[c][n] = xcol[c]
    const float* acol = agg + (size_t)n * BC + bb * C_IN;
    const float* w0row = Wp0 + (otile * 16 + MN) * C_IN;
    const float* w1row = Wp1 + (otile * 16 + MN) * C_IN;

    v8f acc0 = {};
    v8f acc1 = {};
#pragma unroll
    for (int kk = 0; kk < C_IN / 4; ++kk) {
        int c = kk * 4 + k0;
        v2f a0; a0.x = w0row[c]; a0.y = w0row[c + 1];
        v2f bx; bx.x = xcol[c];  bx.y = xcol[c + 1];
        acc0 = __builtin_amdgcn_wmma_f32_16x16x4_f32(
            /*neg_a=*/false, a0, /*neg_b=*/false, bx,
            /*c_mod=*/(short)0, acc0, /*reuse_a=*/false, /*reuse_b=*/false);
        v2f a1; a1.x = w1row[c]; a1.y = w1row[c + 1];
        v2f ba; ba.x = acol[c];  ba.y = acol[c + 1];
        acc1 = __builtin_amdgcn_wmma_f32_16x16x4_f32(
            false, a1, false, ba, (short)0, acc1, false, false);
    }

    float cs = cnt_src[n];
    float as = adj_sum[n];
    float denom = (as == 0.0f) ? 1.0f : as;

    // C/D layout: lane L col N=L%16, VGPR r -> row M = r + (L<16 ? 0 : 8)
    int mbase = (lane < 16) ? 0 : 8;
    float* outb = out + (size_t)bb * 64 * NNODE;
#pragma unroll
    for (int r = 0; r < 8; ++r) {
        int o = otile * 16 + mbase + r;
        float v = (o == 63) ? as
                            : (cs * (acc0[r] + bias_p[o]) + acc1[r]) / denom;
        outb[(size_t)o * NNODE + n] = v;
    }
}

extern "C" void kernel_launch(void* const* d_in, const int* in_sizes, int n_in,
                              void* d_out, int out_size, void* d_ws, size_t ws_size,
                              hipStream_t stream) {
    const float* x       = (const float*)d_in[0];
    const int*   idx     = (const int*)d_in[1];   // (2, E): row0=src, row1=dst
    const float* vals    = (const float*)d_in[2];
    const float* W       = (const float*)d_in[3];
    const float* bvec    = (const float*)d_in[4];
    float* out = (float*)d_out;

    // Workspace layout (floats)
    float* ws = (float*)d_ws;
    float* xt      = ws;                                   // 1,048,576
    float* agg     = xt + (size_t)NNODE * BC;              // 1,048,576
    float* adj_sum = agg + (size_t)NNODE * BC;             // 4096
    float* cnt_src = adj_sum + NNODE;                      // 4096
    float* Wp0     = cnt_src + NNODE;                      // 4096
    float* Wp1     = Wp0 + 64 * C_IN;                      // 4096
    float* bias_p  = Wp1 + 64 * C_IN;                      // 64

    // 1) zero agg + adj_sum + cnt_src (contiguous)
    {
        int nz = NNODE * BC + NNODE + NNODE;
        zero_kernel<<<(nz + 255) / 256, 256, 0, stream>>>(agg, nz);
    }
    // 2) transpose x -> xt
    transpose_kernel<<<(NNODE * BC) / 256, 256, 0, stream>>>(x, xt);
    // 3) pad weights / bias
    padw_kernel<<<(64 * C_IN) / 256, 256, 0, stream>>>(W, bvec, Wp0, Wp1, bias_p);
    // 4) edge stats
    edge_stats_kernel<<<NEDGE / 256, 256, 0, stream>>>(idx, vals, adj_sum, cnt_src);
    // 5) column aggregation (one wave per edge)
    aggregate_kernel<<<(NEDGE * 32) / 256, 256, 0, stream>>>(idx, xt, agg);
    // 6) WMMA GEMM + normalize + counts channel
    {
        int nwaves = BATCH * 4 * (NNODE / 16); // 4096
        gemm_wmma_kernel<<<(nwaves * 32) / 256, 256, 0, stream>>>(
            xt, agg, Wp0, Wp1, bias_p, adj_sum, cnt_src, out);
    }
}